// RelScaledDotProductAttention_69870527971523
// MI455X (gfx1250) — compile-verified
//
#include <hip/hip_runtime.h>

#define NB 16
#define LQ 384
#define LK 384
#define DK 64
#define DV 64

typedef __attribute__((ext_vector_type(2))) float v2f;
typedef __attribute__((ext_vector_type(4))) float v4f;
typedef __attribute__((ext_vector_type(8))) float v8f;

// ---------------------------------------------------------------------------
// Kernel 1: raw scores S = (Q K^T) / 8 via V_WMMA_F32_16X16X4_F32.
// One wave computes one 16x16 tile of S; 4 waves (128 threads) per block.
//
// A (16x4 f32) lane layout: lanes 0-15 hold M=lane, {K=0,K=1}; lanes 16-31
// hold M=lane-16, {K=2,K=3}.  B (4x16) mirrors with N in place of M, and
// since S = Q*K^T, B[k][n] = Kmat[n][k] -> identical addressing into Kmat.
// D (16x16 f32): VGPR v holds M = v + 8*(lane/16), N = lane%16.
// ---------------------------------------------------------------------------
__global__ __launch_bounds__(128) void qk_scores_wmma(
    const float* __restrict__ q, const float* __restrict__ k,
    float* __restrict__ attn)
{
    const int TK = LK / 16;           // 24 tiles along keys
    const int TQ = LQ / 16;           // 24 tiles along queries
    int tileId = blockIdx.x * 4 + (threadIdx.x >> 5);
    int n  = tileId / (TQ * TK);
    int r  = tileId % (TQ * TK);
    int qt = r / TK;
    int kt = r % TK;

    int lane = threadIdx.x & 31;
    int half = lane >> 4;             // 0: K={0,1}+4kb, 1: K={2,3}+4kb
    int mn   = lane & 15;             // M for A, N for B

    const float2* qrow = reinterpret_cast<const float2*>(
        q + (size_t)(n * LQ + qt * 16 + mn) * DK);
    const float2* krow = reinterpret_cast<const float2*>(
        k + (size_t)(n * LK + kt * 16 + mn) * DK);

    v8f acc = {};
#pragma unroll
    for (int kb = 0; kb < 16; ++kb) {         // DK=64 in chunks of 4
        float2 qa = qrow[kb * 2 + half];
        float2 ka = krow[kb * 2 + half];
        v2f a; a[0] = qa.x * 0.125f; a[1] = qa.y * 0.125f;  // fold 1/temperature
        v2f b; b[0] = ka.x;          b[1] = ka.y;
        acc = __builtin_amdgcn_wmma_f32_16x16x4_f32(
            /*neg_a=*/false, a, /*neg_b=*/false, b,
            /*c_mod=*/(short)0, acc, /*reuse_a=*/false, /*reuse_b=*/false);
    }

    float* dst = attn + (size_t)(n * LQ + qt * 16) * LK + kt * 16 + mn;
#pragma unroll
    for (int vv = 0; vv < 8; ++vv) {
        int m = vv + half * 8;
        dst[(size_t)m * LK] = acc[vv];
    }
}

// ---------------------------------------------------------------------------
// Kernel 2: per (n,q) row — softmax of 384 scores in LDS (writes normalized
// attn back = output #2), then stream rel_pos_v[row] (384x64 f32, 98 KB)
// exactly once with NON-TEMPORAL loads (603 MB total, single use, > L2's
// 192 MB — keep it from thrashing L2 so the attn handoff stays resident)
// to produce out[row] (64 f32) = output #1.  Pure-bandwidth kernel.
// ---------------------------------------------------------------------------
__global__ __launch_bounds__(256) void softmax_relv(
    const float* __restrict__ rpv, float* __restrict__ attn,
    float* __restrict__ out)
{
    __shared__ float  s[LK];       // softmax probabilities
    __shared__ float  red[256];    // scalar reduction scratch
    __shared__ v4f    r4[256];     // float4 reduction scratch

    int row = blockIdx.x;          // row = n*LQ + q
    int tid = threadIdx.x;
    float* attnRow = attn + (size_t)row * LK;

    for (int i = tid; i < LK; i += 256) s[i] = attnRow[i];
    __syncthreads();

    // row max
    float m = -3.402823466e38f;
    for (int i = tid; i < LK; i += 256) m = fmaxf(m, s[i]);
    red[tid] = m;
    __syncthreads();
    for (int st = 128; st > 0; st >>= 1) {
        if (tid < st) red[tid] = fmaxf(red[tid], red[tid + st]);
        __syncthreads();
    }
    float rowmax = red[0];
    __syncthreads();

    // exp + sum
    float sum = 0.f;
    for (int i = tid; i < LK; i += 256) {
        float e = __expf(s[i] - rowmax);
        s[i] = e;
        sum += e;
    }
    red[tid] = sum;
    __syncthreads();
    for (int st = 128; st > 0; st >>= 1) {
        if (tid < st) red[tid] += red[tid + st];
        __syncthreads();
    }
    float inv = 1.0f / red[0];
    __syncthreads();

    // normalize + write attn output
    for (int i = tid; i < LK; i += 256) {
        float p = s[i] * inv;
        s[i] = p;
        attnRow[i] = p;
    }
    __syncthreads();

    // weighted sum over k of rel_pos_v: thread (kg, dq) owns d = 4*dq..4*dq+3,
    // k = kg, kg+16, ...  -> 16 threads x float4 = one fully coalesced 256B row.
    int dq = tid & 15;
    int kg = tid >> 4;
    const v4f* rpv4 =
        reinterpret_cast<const v4f*>(rpv + (size_t)row * LK * DV);
    v4f acc = {};
    for (int kk = kg; kk < LK; kk += 16) {
        float p = s[kk];
        v4f vv = __builtin_nontemporal_load(&rpv4[kk * 16 + dq]);  // TH=NT stream
        acc += p * vv;
    }
    r4[tid] = acc;
    __syncthreads();
    for (int st = 8; st > 0; st >>= 1) {
        if (kg < st) {
            r4[tid] = r4[tid] + r4[tid + st * 16];
        }
        __syncthreads();
    }
    if (kg == 0) {
        v4f* outRow = reinterpret_cast<v4f*>(out + (size_t)row * DV);
        outRow[dq] = r4[tid];
    }
}

extern "C" void kernel_launch(void* const* d_in, const int* in_sizes, int n_in,
                              void* d_out, int out_size, void* d_ws, size_t ws_size,
                              hipStream_t stream) {
    (void)in_sizes; (void)n_in; (void)out_size; (void)d_ws; (void)ws_size;
    const float* q   = (const float*)d_in[0];
    const float* k   = (const float*)d_in[1];
    // d_in[2] (v) and d_in[3] (rel_pos) are unused by the reference math.
    const float* rpv = (const float*)d_in[4];

    float* out  = (float*)d_out;                       // [NB, LQ, DV]
    float* attn = out + (size_t)NB * LQ * DV;          // [NB, LQ, LK]

    const int tiles = NB * (LQ / 16) * (LK / 16);      // 9216, /4 waves per block
    qk_scores_wmma<<<tiles / 4, 128, 0, stream>>>(q, k, attn);
    softmax_relv<<<NB * LQ, 256, 0, stream>>>(rpv, attn, out);
}